// InterativeModifier_25898652795269
// MI455X (gfx1250) — compile-verified
//
#include <hip/hip_runtime.h>
#include <stdint.h>

// Iterative raster-scan morphological modifier (Euler-number criterion).
// Strictly sequential reference -> exact skewed-wavefront (d = 2i + j)
// parallelization, bit-packed image resident in LDS, 512-entry fused LUT.

#define IMG_H 512
#define IMG_W 512
#define ROWW 18           // padded 32-bit words per row (1 pad + 16 data + 1 pad)
#define NROWS 514         // padded rows (1 + 512 + 1)
#define NTHREADS 256      // 8 wave32s; max wavefront parallelism is 256
#define STAGE_N 4096      // output staging floats (16 KB)

// 4*EulerNumber (4-connectivity, Gray quad counting) of a 3x3 patch given as
// 9 bits (bit r*3+c). Pads to 5x5 and counts the 16 2x2 quads.
__device__ __forceinline__ int euler4x_9(unsigned m9) {
    unsigned long long b = 0ull;
#pragma unroll
    for (int r = 0; r < 3; ++r)
#pragma unroll
        for (int c = 0; c < 3; ++c)
            if ((m9 >> (r * 3 + c)) & 1u)
                b |= 1ull << ((r + 1) * 5 + (c + 1));
    int q1 = 0, q3 = 0, qd = 0;
#pragma unroll
    for (int r = 0; r < 4; ++r)
#pragma unroll
        for (int c = 0; c < 4; ++c) {
            int a  = (int)((b >> (r * 5 + c)) & 1ull);
            int cc = (int)((b >> (r * 5 + c + 1)) & 1ull);
            int dd = (int)((b >> ((r + 1) * 5 + c)) & 1ull);
            int e  = (int)((b >> ((r + 1) * 5 + c + 1)) & 1ull);
            int s = a + cc + dd + e;
            q1 += (s == 1);
            q3 += (s == 3);
            qd += ((s == 2) && (a == e));
        }
    return q1 - q3 + 2 * qd;
}

// Read 3 consecutive bits (cols j-1..j+1) from a padded bitmap row.
// rowp points at the row's pad word; col j lives at word 1 + j/32.
__device__ __forceinline__ unsigned read3(const unsigned* rowp, int j) {
    int base = j - 1;                 // in [-1, 510]
    int wb = 1 + (base >> 5);         // arithmetic shift: base=-1 -> word 0 (pad)
    int ob = base & 31;
    unsigned long long w =
        ((unsigned long long)rowp[wb + 1] << 32) | (unsigned long long)rowp[wb];
    return (unsigned)((w >> ob) & 7ull);
}

__global__ __launch_bounds__(NTHREADS)
void InterativeModifier_25898652795269_kernel(const float* __restrict__ x,
                                              const float* __restrict__ op,
                                              float* __restrict__ out) {
    __shared__ unsigned bitmap[NROWS * ROWW];   // ~37 KB packed image state
    __shared__ unsigned lut[16];                // 512-entry fused update LUT
    __shared__ __align__(16) float stage[STAGE_N];
    const int t = (int)threadIdx.x;

    // ---- Phase 0: zero padded bitmap and LUT ----
    for (int k = t; k < NROWS * ROWW; k += NTHREADS) bitmap[k] = 0u;
    if (t < 16) lut[t] = 0u;
    __syncthreads();

    // ---- Phase 1: fuse the whole update rule into a 512-entry bit LUT ----
    // cfg = 9 patch bits (bit 4 = center). new = operation[2*(dE==4) + center].
    for (int cfg = t; cfg < 512; cfg += NTHREADS) {
        int e_before = euler4x_9((unsigned)cfg);
        int e_after  = euler4x_9((unsigned)cfg ^ 16u);  // flip center
        int aij = ((e_after - e_before) == 4) ? 1 : 0;  // NEIGHBOR_TYPE == 1
        int bij = (cfg >> 4) & 1;
        unsigned bit = (op[2 * aij + bij] > 0.5f) ? 1u : 0u;
        if (bit) atomicOr(&lut[cfg >> 5], 1u << (cfg & 31));
    }

    // ---- Phase 2: binarize + bit-pack input. wave32 ballot = 1 bitmap word ----
    for (int p = t; p < IMG_H * IMG_W; p += NTHREADS) {
        int row = p >> 9;
        int col = p & (IMG_W - 1);
        unsigned mask = __builtin_amdgcn_ballot_w32(x[p] > 0.5f);
        if ((t & 31) == 0)                       // lane 0: col is 32-aligned
            bitmap[(row + 1) * ROWW + 1 + (col >> 5)] = mask;
    }
    __syncthreads();

    // ---- Phase 3: skewed wavefront d = 2i + j (exactly reproduces the
    // sequential raster-scan semantics; 1534 dependent steps). ----
    const int DMAX = 2 * (IMG_H - 1) + (IMG_W - 1);   // 1533
    for (int d = 0; d <= DMAX; ++d) {
        int ilo = (d > (IMG_W - 1)) ? ((d - (IMG_W - 1) + 1) >> 1) : 0;
        int i = ilo + t;
        int j = d - 2 * i;
        if (i < IMG_H && j >= 0) {               // j <= 511 guaranteed by ilo
            const unsigned* rtop = &bitmap[(i    ) * ROWW];
            unsigned*       rmid = &bitmap[(i + 1) * ROWW];
            const unsigned* rbot = &bitmap[(i + 2) * ROWW];
            unsigned idx9 = read3(rtop, j) | (read3(rmid, j) << 3) |
                            (read3(rbot, j) << 6);
            unsigned bnew = (lut[idx9 >> 5] >> (idx9 & 31)) & 1u;
            unsigned* wp = &rmid[1 + (j >> 5)];
            unsigned m = 1u << (j & 31);
            unsigned old = *wp;                  // no same-step conflicts (proved
            *wp = bnew ? (old | m) : (old & ~m); // at bit & word granularity)
        }
        __syncthreads();
    }

    // ---- Phase 4: unpack to floats in LDS, stream out with async stores ----
    for (int base = 0; base < IMG_H * IMG_W; base += STAGE_N) {
#pragma unroll
        for (int s = 0; s < 16; ++s) {
            int q = t * 16 + s;
            int p = base + q;
            int row = p >> 9, col = p & (IMG_W - 1);
            unsigned w = bitmap[(row + 1) * ROWW + 1 + (col >> 5)];
            stage[q] = (float)((w >> (col & 31)) & 1u);
        }
        asm volatile("s_wait_dscnt 0" ::: "memory");   // LDS writes visible to TDM path
#pragma unroll
        for (int s = 0; s < 4; ++s) {
            int q = t * 16 + s * 4;                    // 16B-aligned LDS + global
            const float* gp = out + base + q;
            unsigned loff = (unsigned)(uintptr_t)&stage[q];  // low 32b = LDS offset
            asm volatile("global_store_async_from_lds_b128 %0, %1, off"
                         :: "v"(gp), "v"(loff) : "memory");
        }
        asm volatile("s_wait_asynccnt 0" ::: "memory"); // before stage reuse
    }
}

extern "C" void kernel_launch(void* const* d_in, const int* in_sizes, int n_in,
                              void* d_out, int out_size, void* d_ws, size_t ws_size,
                              hipStream_t stream) {
    (void)in_sizes; (void)n_in; (void)out_size; (void)d_ws; (void)ws_size;
    const float* x  = (const float*)d_in[0];   // (1, 512, 512) float32
    const float* op = (const float*)d_in[1];   // (4,) float32
    float* out = (float*)d_out;                // (1, 512, 512) float32
    // Single workgroup: the dependence chain forces one cooperating WGP; all
    // mutable state lives in its 320 KB LDS, HBM is touched only at the ends.
    InterativeModifier_25898652795269_kernel<<<dim3(1), dim3(NTHREADS), 0, stream>>>(x, op, out);
}